// HomomorphismAutoencoder_6940667150755
// MI455X (gfx1250) — compile-verified
//
#include <hip/hip_runtime.h>

typedef float v2f __attribute__((ext_vector_type(2)));
typedef float v8f __attribute__((ext_vector_type(8)));

#define B_   64
#define T_   256
#define TA_  255
#define OBS_ 512
#define ACT_ 32
#define LAT_ 64
#define PHI_ 512

// ---------------------------------------------------------------------------
// fp32 WMMA GEMM: C[M,N] = A[M,K] @ W[K,N] + bias[N]
// Compile-time K/N/NT: immediate-offset loads + pointer bump (no per-iter
// 64-bit muls). One wave per 16x(16*NT) output strip: the A fragment (one
// 8-byte load per lane per k-step) feeds NT v_wmma_f32_16x16x4_f32 ops.
// A fragment layout (ISA 7.12.2, 32-bit A 16x4):
//   lanes 0-15: M=lane, VGPR0=K0, VGPR1=K1 ; lanes 16-31: M=lane-16, K2/K3
// B fragment: rows striped across lanes within a VGPR (VGPR r = row r / r+2).
// ---------------------------------------------------------------------------
template <int K, int N, int NT>
__global__ void wmma_gemm_f32(const float* __restrict__ A,
                              const float* __restrict__ W,
                              const float* __restrict__ bias,
                              float* __restrict__ C) {
  const int lane = threadIdx.x;     // blockDim.x == 32 (one wave, EXEC all-1)
  const int half = lane >> 4;
  const int l    = lane & 15;
  const int m0 = blockIdx.x << 4;
  const int n0 = blockIdx.y * (16 * NT);

  const float* Ap = A + (size_t)(m0 + l) * K + (half << 1);
  const float* Bp = W + (size_t)(half << 1) * N + n0 + l;

  v8f acc[NT];
#pragma unroll
  for (int nt = 0; nt < NT; ++nt)
    acc[nt] = (v8f){0.f, 0.f, 0.f, 0.f, 0.f, 0.f, 0.f, 0.f};

  for (int k0 = 0; k0 < K; k0 += 4) {
    if (K >= 256 && (k0 & 63) == 0)          // stream-ahead hint for A rows
      __builtin_prefetch(Ap + 64, 0, 3);     // -> global_prefetch_b8 (near)
    v2f a;
    a.x = Ap[0];
    a.y = Ap[1];
#pragma unroll
    for (int nt = 0; nt < NT; ++nt) {
      v2f b;
      b.x = Bp[nt * 16];                     // row kb   = k0 + 2*half
      b.y = Bp[nt * 16 + N];                 // row kb+1
      acc[nt] = __builtin_amdgcn_wmma_f32_16x16x4_f32(
          false, a, false, b, (short)0, acc[nt], false, false);
    }
    Ap += 4;
    Bp += 4 * N;
  }

  float bv[NT];
#pragma unroll
  for (int nt = 0; nt < NT; ++nt) bv[nt] = bias[n0 + nt * 16 + l];

  float* Crow = C + (size_t)(m0 + (half << 3)) * N + n0 + l;
#pragma unroll
  for (int r = 0; r < 8; ++r)
#pragma unroll
    for (int nt = 0; nt < NT; ++nt)
      Crow[(size_t)r * N + nt * 16] = acc[nt][r] + bv[nt];
}

// ---------------------------------------------------------------------------
// Per-(b,t): transpose the 8 8x8 blocks of after_phi, expm each block
// (uniform scaling-and-squaring: X/2^6, 10-term Taylor, 6 squarings),
// then emit the full 64x64 block-diagonal rho tile with coalesced b128 stores.
// Workgroup = 64 threads = 8 blocks x 8 rows.
// ---------------------------------------------------------------------------
__global__ void expm_rho_kernel(const float* __restrict__ after_phi,
                                float* __restrict__ rho) {
  __shared__ float X[8][8][8];   // [block][i][j], pre-scaled by 2^-6
  __shared__ float P[8][8][8];
  const int tid = threadIdx.x;   // 0..63
  const int k = tid >> 3;
  const int i = tid & 7;
  const float* g = after_phi + (size_t)blockIdx.x * PHI_;

  // blocks = reshape(8,8).T  =>  X[i][j] = after_phi[k*64 + j*8 + i]
#pragma unroll
  for (int j = 0; j < 8; ++j)
    X[k][i][j] = 0.015625f * g[k * 64 + j * 8 + i];
  __syncthreads();

  float Pr[8], Tr[8];
#pragma unroll
  for (int j = 0; j < 8; ++j) {
    float x = X[k][i][j];
    Tr[j] = x;
    Pr[j] = x + (j == i ? 1.0f : 0.0f);
  }
  // Taylor: P = sum_{n<=10} Xs^n / n!
  for (int n = 2; n <= 10; ++n) {
    const float inv = 1.0f / (float)n;
    float nt[8];
#pragma unroll
    for (int j = 0; j < 8; ++j) {
      float s = 0.f;
#pragma unroll
      for (int kk = 0; kk < 8; ++kk) s += Tr[kk] * X[k][kk][j];
      nt[j] = s * inv;
    }
#pragma unroll
    for (int j = 0; j < 8; ++j) { Tr[j] = nt[j]; Pr[j] += nt[j]; }
  }
  // 6 squarings (uniform count -> barrier-safe)
  for (int sq = 0; sq < 6; ++sq) {
#pragma unroll
    for (int j = 0; j < 8; ++j) P[k][i][j] = Pr[j];
    __syncthreads();
    float np[8];
#pragma unroll
    for (int j = 0; j < 8; ++j) {
      float s = 0.f;
#pragma unroll
      for (int kk = 0; kk < 8; ++kk) s += Pr[kk] * P[k][kk][j];
      np[j] = s;
    }
    __syncthreads();
#pragma unroll
    for (int j = 0; j < 8; ++j) Pr[j] = np[j];
  }
#pragma unroll
  for (int j = 0; j < 8; ++j) P[k][i][j] = Pr[j];
  __syncthreads();

  // Coalesced write of the full 64x64 tile (1024 float4, 16 per thread).
  float4* dst = (float4*)(rho + (size_t)blockIdx.x * (LAT_ * LAT_));
#pragma unroll
  for (int q = 0; q < 16; ++q) {
    const int idx4 = tid + (q << 6);
    const int row  = idx4 >> 4;
    const int c4   = idx4 & 15;
    const int kb   = row >> 3;
    float4 v = make_float4(0.f, 0.f, 0.f, 0.f);
    if (c4 == (kb << 1)) {
      v = make_float4(P[kb][row & 7][0], P[kb][row & 7][1],
                      P[kb][row & 7][2], P[kb][row & 7][3]);
    } else if (c4 == (kb << 1) + 1) {
      v = make_float4(P[kb][row & 7][4], P[kb][row & 7][5],
                      P[kb][row & 7][6], P[kb][row & 7][7]);
    }
    dst[idx4] = v;
  }
}

// ---------------------------------------------------------------------------
// Sequential scan: block-diagonal rho => 512 independent 8-dim chains.
// Thread (b,k): h <- h @ expm_block_k(t), t = 0..254.
// ---------------------------------------------------------------------------
__global__ void scan_kernel(const float* __restrict__ latent_obs,
                            const float* __restrict__ rho,
                            float* __restrict__ pred_lat) {
  const int idx = blockIdx.x * blockDim.x + threadIdx.x;
  if (idx >= B_ * 8) return;
  const int b = idx >> 3;
  const int k = idx & 7;

  float h[8];
  const float* h0 = latent_obs + (size_t)b * T_ * LAT_ + k * 8;
  float* out = pred_lat + (size_t)b * T_ * LAT_ + k * 8;
#pragma unroll
  for (int j = 0; j < 8; ++j) { h[j] = h0[j]; out[j] = h[j]; }

  for (int t = 0; t < TA_; ++t) {
    const float* Mb =
        rho + ((size_t)(b * TA_ + t) * LAT_ + k * 8) * LAT_ + k * 8;
    float hn[8] = {0.f, 0.f, 0.f, 0.f, 0.f, 0.f, 0.f, 0.f};
#pragma unroll
    for (int i = 0; i < 8; ++i) {
      const float4 m0 = *(const float4*)(Mb + (size_t)i * LAT_);
      const float4 m1 = *(const float4*)(Mb + (size_t)i * LAT_ + 4);
      hn[0] += h[i] * m0.x; hn[1] += h[i] * m0.y;
      hn[2] += h[i] * m0.z; hn[3] += h[i] * m0.w;
      hn[4] += h[i] * m1.x; hn[5] += h[i] * m1.y;
      hn[6] += h[i] * m1.z; hn[7] += h[i] * m1.w;
    }
    float* o = out + (size_t)(t + 1) * LAT_;
#pragma unroll
    for (int j = 0; j < 8; ++j) { h[j] = hn[j]; o[j] = hn[j]; }
  }
}

// ---------------------------------------------------------------------------
extern "C" void kernel_launch(void* const* d_in, const int* in_sizes, int n_in,
                              void* d_out, int out_size, void* d_ws,
                              size_t ws_size, hipStream_t stream) {
  const float* obs   = (const float*)d_in[0];
  const float* act   = (const float*)d_in[1];
  const float* W_enc = (const float*)d_in[2];
  const float* b_enc = (const float*)d_in[3];
  const float* W_dec = (const float*)d_in[4];
  const float* b_dec = (const float*)d_in[5];
  const float* W_phi = (const float*)d_in[6];
  const float* b_phi = (const float*)d_in[7];

  float* out        = (float*)d_out;
  float* latent_obs = out;                                   // 16384*64
  float* rho        = out + 1048576;                         // 16320*4096
  float* pred_lat   = out + 1048576 + 66846720;              // 16384*64
  float* pred_obs   = out + 1048576 + 66846720 + 1048576;    // 16384*512

  float* after_phi = (float*)d_ws;                           // 16320*512 f32

  // 1) encoder: (16384 x 512) @ (512 x 64); NT=4 covers all of N=64
  wmma_gemm_f32<OBS_, LAT_, 4>
      <<<dim3(1024, 1), 32, 0, stream>>>(obs, W_enc, b_enc, latent_obs);
  // 2) phi: (16320 x 32) @ (32 x 512)
  wmma_gemm_f32<ACT_, PHI_, 4>
      <<<dim3(1020, 8), 32, 0, stream>>>(act, W_phi, b_phi, after_phi);
  // 3) expm of 8x8 blocks + block-diagonal rho fill (one streaming pass)
  expm_rho_kernel<<<16320, 64, 0, stream>>>(after_phi, rho);
  // 4) latent scan (512 independent chains)
  scan_kernel<<<8, 64, 0, stream>>>(latent_obs, rho, pred_lat);
  // 5) decoder: (16384 x 64) @ (64 x 512)
  wmma_gemm_f32<LAT_, OBS_, 4>
      <<<dim3(1024, 8), 32, 0, stream>>>(pred_lat, W_dec, b_dec, pred_obs);
}